// Random_Augmentation_Dasp_44890998178605
// MI455X (gfx1250) — compile-verified
//
#include <hip/hip_runtime.h>
#include <cstdint>

// ---------------------------------------------------------------------------
// Multiband dynamics chain for MI455X (gfx1250, wave32).
//  - LR4 crossover implemented as exact blocked state-space IIR:
//      per 16-sample block:  Y = H(16x16)*X + Obs(16x4)*s,  s' = A16*s + K*X
//    H*X and Kpad*X are done with V_WMMA_F32_16X16X4_F32 (fp32 tensor op).
//  - Nonlinear attack/release smoothers are sequential per sequence
//    (data-dependent coefficient), parallel across 48+16 sequences.
// ---------------------------------------------------------------------------

#define T_LEN   131072
#define BSZ     16
#define CHS     2
#define NSEQ_F  64      // 2 filters * 16 batch * 2 ch
#define NBLK    8192    // T/16
#define NGRP    512     // T/256

typedef float v2f __attribute__((ext_vector_type(2)));
typedef float v8f __attribute__((ext_vector_type(8)));

// ---- workspace offsets (in floats) ----------------------------------------
static constexpr size_t OFF_Y0   = 0;                                    // [16][2][T]
static constexpr size_t OFF_LOW  = OFF_Y0  + (size_t)BSZ*CHS*T_LEN;      // [32][T]
static constexpr size_t OFF_HIGH = OFF_LOW + (size_t)BSZ*CHS*T_LEN;      // [32][T]
static constexpr size_t OFF_U    = OFF_HIGH+ (size_t)BSZ*CHS*T_LEN;      // [64][8192][4]
static constexpr size_t OFF_S    = OFF_U   + (size_t)NSEQ_F*NBLK*4;      // [64][8192][4]
static constexpr size_t OFF_G    = OFF_S   + (size_t)NSEQ_F*NBLK*4;      // [48][T]
static constexpr size_t OFF_DET  = OFF_G   + (size_t)48*T_LEN;           // [16][T]
static constexpr size_t OFF_ENV  = OFF_DET + (size_t)BSZ*T_LEN;          // [16][T]
static constexpr size_t OFF_PAR  = OFF_ENV + (size_t)BSZ*T_LEN;
// total ~= 27.3 M floats ~= 110 MB (fits comfortably in MI455X's 192 MB L2)

// ---- param area, offsets relative to par base -----------------------------
enum : int {
  RP_HMAT = 0,          // [2][16][16][16]  Toeplitz of h[0..15]
  RP_KPAD = 8192,       // [2][16][16][16]  rows0-3 = K(4x16), rest 0
  RP_A16  = 16384,      // [2][16][4][4]    16-step state transition
  RP_OBS  = 16896,      // [2][16][16][4]   observability (zero-input outputs)
  RP_DRV  = 18944,      // [16] linear drive gain
  RP_AT48 = 18960, RP_RT48 = 19008,
  RP_C148 = 19056, RP_CT48 = 19104,
  RP_E148 = 19152, RP_ET48 = 19200,
  RP_ATL  = 19248, RP_RTL = 19264, RP_LTL = 19280
};

struct BQ { float b0, b1, b2, a1, a2; };

// DF2T biquad cascade (LR4 = same biquad applied twice), 4 state vars
__device__ __forceinline__ float lr4_step(const BQ& q, float s[4], float x) {
  float y1 = q.b0 * x + s[0];
  s[0] = q.b1 * x - q.a1 * y1 + s[1];
  s[1] = q.b2 * x - q.a2 * y1;
  float y2 = q.b0 * y1 + s[2];
  s[2] = q.b1 * y1 - q.a1 * y2 + s[3];
  s[3] = q.b2 * y1 - q.a2 * y2;
  return y2;
}

__device__ __forceinline__ float ms2coef_d(float ms, float sr) {
  return 1.0f - expf(-2200.0f / (ms * sr));
}

// ---------------------------------------------------------------------------
// Kernel 1: derive all per-batch parameters + blocked state-space matrices
// ---------------------------------------------------------------------------
__global__ void k_coeffs(const float* __restrict__ drive_db,
                         const float* __restrict__ low_cut,
                         const float* __restrict__ high_cut,
                         const float* __restrict__ cth, const float* __restrict__ cra,
                         const float* __restrict__ eth, const float* __restrict__ era,
                         const float* __restrict__ atm, const float* __restrict__ rtm,
                         const float* __restrict__ lth, const float* __restrict__ lat,
                         const float* __restrict__ lrt,
                         const int* __restrict__ sr_i,
                         float* __restrict__ par) {
  const int tid = threadIdx.x;           // 64 threads, 1 block
  const float sr = (float)(*sr_i);
  if (tid < 32) {                        // (filter, batch) pairs
    const int f = tid >> 4, b = tid & 15, fb = tid;
    const float cut = (f == 0) ? low_cut[b] : high_cut[b];
    const float w0 = 6.28318530718f * cut / sr;
    const float sw = sinf(w0), cw = cosf(w0);
    const float alpha = sw * 0.70710678f;
    const float a0 = 1.0f + alpha;
    BQ q;
    if (f == 0) { q.b0 = (1.f - cw) * 0.5f / a0; q.b1 = (1.f - cw) / a0; q.b2 = q.b0; }
    else        { q.b0 = (1.f + cw) * 0.5f / a0; q.b1 = -(1.f + cw) / a0; q.b2 = q.b0; }
    q.a1 = -2.f * cw / a0; q.a2 = (1.f - alpha) / a0;

    float* H = par + RP_HMAT + fb * 256;
    float* K = par + RP_KPAD + fb * 256;
    float* A = par + RP_A16  + fb * 16;
    float* O = par + RP_OBS  + fb * 64;

    float h[16]; float s[4] = {0, 0, 0, 0};
    for (int t = 0; t < 16; ++t) h[t] = lr4_step(q, s, (t == 0) ? 1.f : 0.f);
    for (int r = 0; r < 16; ++r)
      for (int c = 0; c < 16; ++c)
        H[r * 16 + c] = (r >= c) ? h[r - c] : 0.f;

    for (int i = 0; i < 4; ++i) {        // zero-input responses of basis states
      float ss[4] = {0, 0, 0, 0}; ss[i] = 1.f;
      for (int t = 0; t < 16; ++t) O[t * 4 + i] = lr4_step(q, ss, 0.f);
      for (int r = 0; r < 4; ++r) A[r * 4 + i] = ss[r];
    }
    for (int c = 0; c < 16; ++c) {       // end-of-block state from impulse at c
      float ss[4] = {0, 0, 0, 0};
      for (int t = 0; t < 16; ++t) (void)lr4_step(q, ss, (t == c) ? 1.f : 0.f);
      for (int r = 0; r < 4; ++r) K[r * 16 + c] = ss[r];
    }
    for (int r = 4; r < 16; ++r)
      for (int c = 0; c < 16; ++c) K[r * 16 + c] = 0.f;
  }
  if (tid < 48) {                        // seq = band*16 + b  (matches ct.T.reshape(-1))
    const int band = tid >> 4, b = tid & 15, src = b * 3 + band;
    par[RP_AT48 + tid] = ms2coef_d(atm[src], sr);
    par[RP_RT48 + tid] = ms2coef_d(rtm[src], sr);
    par[RP_C148 + tid] = 1.f - 1.f / cra[src];
    par[RP_CT48 + tid] = cth[src];
    par[RP_E148 + tid] = 1.f - 1.f / era[src];
    par[RP_ET48 + tid] = eth[src];
  }
  if (tid < 16) {
    par[RP_DRV + tid] = exp2f(drive_db[tid] * 0.16609640474f);   // 10^(db/20)
    par[RP_ATL + tid] = ms2coef_d(lat[tid], sr);
    par[RP_RTL + tid] = ms2coef_d(lrt[tid], sr);
    par[RP_LTL + tid] = exp2f(lth[tid] * 0.16609640474f);
  }
}

// ---------------------------------------------------------------------------
// Kernel 2: drive distortion (elementwise, bandwidth-bound)
// ---------------------------------------------------------------------------
__global__ void k_dist(const float* __restrict__ x, const float* __restrict__ dist_mix,
                       const float* __restrict__ par, float* __restrict__ y0) {
  const size_t idx = (size_t)blockIdx.x * blockDim.x + threadIdx.x;
  if (idx >= (size_t)BSZ * CHS * T_LEN) return;
  const int b = (int)(idx / ((size_t)CHS * T_LEN));
  const float w = dist_mix[b];
  const float gd = par[RP_DRV + b];
  const float xv = x[idx];
  y0[idx] = w * tanhf(xv * gd) + (1.0f - w) * xv;
}

// ---------------------------------------------------------------------------
// Kernel 3: zero-state block FIR (Y = H*X) + block drive terms (U = Kpad*X)
// via V_WMMA_F32_16X16X4_F32.  One wave per block; EXEC all-1s at the WMMAs.
// grid = (64 sequences, 16 group-slices), block = 32.
// ---------------------------------------------------------------------------
__global__ void k_fir(const float* __restrict__ y0, const float* __restrict__ par,
                      float* __restrict__ low, float* __restrict__ high,
                      float* __restrict__ ubuf) {
  const int lane = threadIdx.x;            // 0..31
  const int seq  = blockIdx.x;             // 0..63: f*32 + b*2 + ch
  const int f    = seq >> 5;
  const int bc   = seq & 31;
  const int b    = bc >> 1;
  const int fb   = f * 16 + b;
  const float* xseq = y0 + (size_t)bc * T_LEN;
  float* yout = (f == 0 ? low : high) + (size_t)bc * T_LEN;
  float* useq = ubuf + (size_t)seq * NBLK * 4;

  const int m   = lane & 15;               // A-matrix row / B,N column / D column
  const int hiH = lane >> 4;               // lane half select

  // A fragments: VGPR0 = K0 (lanes0-15) / K2 (lanes16-31), VGPR1 = K1/K3
  v2f aH[4], aK[4];
  const float* Hrow = par + RP_HMAT + fb * 256 + m * 16;
  const float* Krow = par + RP_KPAD + fb * 256 + m * 16;
#pragma unroll
  for (int k = 0; k < 4; ++k) {
    const int c0 = 4 * k + (hiH ? 2 : 0);
    aH[k].x = Hrow[c0]; aH[k].y = Hrow[c0 + 1];
    aK[k].x = Krow[c0]; aK[k].y = Krow[c0 + 1];
  }

  const int g0 = blockIdx.y * 32;          // 32 groups of 16 blocks each
  for (int gi = 0; gi < 32; ++gi) {
    const int t0 = (g0 + gi) * 256;
    __builtin_prefetch(xseq + t0 + 256, 0, 0);   // global_prefetch for next group
    v8f accY = {};
    v8f accU = {};
#pragma unroll
    for (int k = 0; k < 4; ++k) {
      // X[r][c] = xseq[t0 + c*16 + r]; this lane holds column c = m,
      // rows {4k,4k+1} (lanes 0-15) or {4k+2,4k+3} (lanes 16-31).
      const float4 qv = *reinterpret_cast<const float4*>(xseq + t0 + m * 16 + 4 * k);
      v2f bk;
      bk.x = hiH ? qv.z : qv.x;
      bk.y = hiH ? qv.w : qv.y;
      accY = __builtin_amdgcn_wmma_f32_16x16x4_f32(false, aH[k], false, bk,
                                                   (short)0, accY, false, false);
      accU = __builtin_amdgcn_wmma_f32_16x16x4_f32(false, aK[k], false, bk,
                                                   (short)0, accU, false, false);
    }
    // store zero-state outputs: D vgpr v -> (M = v + 8*hiH, N = m)
    const int rowM0 = hiH ? 8 : 0;
#pragma unroll
    for (int v = 0; v < 8; ++v)
      yout[t0 + m * 16 + rowM0 + v] = accY[v];
    // store block drive u (rows 0..3 live in accU[0..3], lanes 0-15, N = block)
    if (!hiH) {
      float4 uq; uq.x = accU[0]; uq.y = accU[1]; uq.z = accU[2]; uq.w = accU[3];
      *reinterpret_cast<float4*>(useq + (size_t)((t0 >> 4) + m) * 4) = uq;
    }
  }
}

// ---------------------------------------------------------------------------
// Kernel 4: block-state scan  s_{j+1} = A16 * s_j + u_j   (8192 steps/seq)
// ---------------------------------------------------------------------------
__global__ void k_scan(const float* __restrict__ par, const float* __restrict__ ubuf,
                       float* __restrict__ sbuf) {
  const int seq = blockIdx.x * blockDim.x + threadIdx.x;
  if (seq >= NSEQ_F) return;
  const int f = seq >> 5, b = (seq >> 1) & 15, fb = f * 16 + b;
  const float* A = par + RP_A16 + fb * 16;
  float a[16];
#pragma unroll
  for (int i = 0; i < 16; ++i) a[i] = A[i];
  const float* u = ubuf + (size_t)seq * NBLK * 4;
  float* s = sbuf + (size_t)seq * NBLK * 4;
  float s0 = 0.f, s1 = 0.f, s2 = 0.f, s3 = 0.f;
  for (int j = 0; j < NBLK; ++j) {
    float4 sv; sv.x = s0; sv.y = s1; sv.z = s2; sv.w = s3;
    *reinterpret_cast<float4*>(s + (size_t)j * 4) = sv;      // state at block start
    const float4 uv = *reinterpret_cast<const float4*>(u + (size_t)j * 4);
    const float n0 = a[0]  * s0 + a[1]  * s1 + a[2]  * s2 + a[3]  * s3 + uv.x;
    const float n1 = a[4]  * s0 + a[5]  * s1 + a[6]  * s2 + a[7]  * s3 + uv.y;
    const float n2 = a[8]  * s0 + a[9]  * s1 + a[10] * s2 + a[11] * s3 + uv.z;
    const float n3 = a[12] * s0 + a[13] * s1 + a[14] * s2 + a[15] * s3 + uv.w;
    s0 = n0; s1 = n1; s2 = n2; s3 = n3;
  }
}

// ---------------------------------------------------------------------------
// Kernel 5: add zero-input response:  y[t] += Obs[t%16] . s_block
// ---------------------------------------------------------------------------
__global__ void k_combine(const float* __restrict__ par, const float* __restrict__ sbuf,
                          float* __restrict__ low, float* __restrict__ high) {
  const size_t idx = (size_t)blockIdx.x * blockDim.x + threadIdx.x;
  if (idx >= (size_t)NSEQ_F * T_LEN) return;
  const int seq = (int)(idx / T_LEN);
  const int t   = (int)(idx % T_LEN);
  const int f = seq >> 5, b = (seq >> 1) & 15, fb = f * 16 + b;
  const int r = t & 15, blk = t >> 4;
  const float* O  = par + RP_OBS + fb * 64 + r * 4;
  const float* sv = sbuf + (size_t)seq * NBLK * 4 + (size_t)blk * 4;
  float* buf = (f == 0 ? low : high);
  const size_t p = (size_t)(seq & 31) * T_LEN + t;
  buf[p] += O[0] * sv[0] + O[1] * sv[1] + O[2] * sv[2] + O[3] * sv[3];
}

// ---------------------------------------------------------------------------
// Kernel 6: detectors + static gain curve  g = 10^(g_db/20)
// ---------------------------------------------------------------------------
__global__ void k_detgain(const float* __restrict__ y0, const float* __restrict__ low,
                          const float* __restrict__ high, const float* __restrict__ par,
                          float* __restrict__ g) {
  const size_t idx = (size_t)blockIdx.x * blockDim.x + threadIdx.x;
  if (idx >= (size_t)48 * T_LEN) return;
  const int t = (int)(idx % T_LEN);
  const int s = (int)(idx / T_LEN);     // s = band*16 + b
  const int band = s >> 4, b = s & 15;
  const size_t p0 = (size_t)(b * 2 + 0) * T_LEN + t, p1 = p0 + T_LEN;
  float v;
  if (band == 0)      v = low[p0] + low[p1];
  else if (band == 2) v = high[p0] + high[p1];
  else                v = (y0[p0] - low[p0] - high[p0]) + (y0[p1] - low[p1] - high[p1]);
  const float det = fabsf(v);
  const float xdb = 6.02059991f * log2f(fmaxf(det, 1e-7f));       // 20*log10
  const float gdb = fminf(fminf(par[RP_C148 + s] * (par[RP_CT48 + s] - xdb),
                                par[RP_E148 + s] * (par[RP_ET48 + s] - xdb)), 0.0f);
  g[idx] = exp2f(gdb * 0.16609640474f);                           // 10^(gdb/20)
}

// ---------------------------------------------------------------------------
// Kernel 7: compressor gain smoother (attack_on_rise = False): 48 sequences
// ---------------------------------------------------------------------------
__global__ void k_smooth_comp(const float* __restrict__ par, float* __restrict__ g) {
  const int s = blockIdx.x * blockDim.x + threadIdx.x;
  if (s >= 48) return;
  const float at = par[RP_AT48 + s], rt = par[RP_RT48 + s];
  float* gp = g + (size_t)s * T_LEN;
  float fv = 1.0f;
  for (int t = 0; t < T_LEN; t += 4) {
    float4 q = *reinterpret_cast<float4*>(gp + t);
    float c;
    c = (q.x < fv) ? at : rt; fv += c * (q.x - fv); q.x = fv;
    c = (q.y < fv) ? at : rt; fv += c * (q.y - fv); q.y = fv;
    c = (q.z < fv) ? at : rt; fv += c * (q.z - fv); q.z = fv;
    c = (q.w < fv) ? at : rt; fv += c * (q.w - fv); q.w = fv;
    *reinterpret_cast<float4*>(gp + t) = q;
  }
}

// ---------------------------------------------------------------------------
// Kernel 8: apply band gains + mb mix (in place on y0), limiter detector
// ---------------------------------------------------------------------------
__global__ void k_mbmix(const float* __restrict__ low, const float* __restrict__ high,
                        const float* __restrict__ g, const float* __restrict__ mb_mix,
                        float* __restrict__ y0, float* __restrict__ det) {
  const size_t idx = (size_t)blockIdx.x * blockDim.x + threadIdx.x;
  if (idx >= (size_t)BSZ * T_LEN) return;
  const int b = (int)(idx / T_LEN);
  const int t = (int)(idx % T_LEN);
  const float g0 = g[(size_t)(b)      * T_LEN + t];
  const float g1 = g[(size_t)(16 + b) * T_LEN + t];
  const float g2 = g[(size_t)(32 + b) * T_LEN + t];
  const float w = mb_mix[b];
  float acc = 0.f;
#pragma unroll
  for (int ch = 0; ch < 2; ++ch) {
    const size_t p = (size_t)(b * 2 + ch) * T_LEN + t;
    const float lo = low[p], hi = high[p], xv = y0[p];
    const float mid = xv - lo - hi;
    const float y = lo * g0 + mid * g1 + hi * g2;
    const float out = w * y + (1.f - w) * xv;
    y0[p] = out;
    acc += out;
  }
  det[idx] = fabsf(acc);
}

// ---------------------------------------------------------------------------
// Kernel 9: limiter envelope smoother (attack_on_rise = True): 16 sequences
// ---------------------------------------------------------------------------
__global__ void k_smooth_lim(const float* __restrict__ par, const float* __restrict__ det,
                             float* __restrict__ env) {
  const int b = threadIdx.x;
  if (b >= 16) return;
  const float at = par[RP_ATL + b], rt = par[RP_RTL + b];
  const float* dp = det + (size_t)b * T_LEN;
  float* ep = env + (size_t)b * T_LEN;
  float fv = 0.0f;
  for (int t = 0; t < T_LEN; t += 4) {
    const float4 q = *reinterpret_cast<const float4*>(dp + t);
    float4 o; float c;
    c = (q.x > fv) ? at : rt; fv += c * (q.x - fv); o.x = fv;
    c = (q.y > fv) ? at : rt; fv += c * (q.y - fv); o.y = fv;
    c = (q.z > fv) ? at : rt; fv += c * (q.z - fv); o.z = fv;
    c = (q.w > fv) ? at : rt; fv += c * (q.w - fv); o.w = fv;
    *reinterpret_cast<float4*>(ep + t) = o;
  }
}

// ---------------------------------------------------------------------------
// Kernel 10: limiter gain + output
// ---------------------------------------------------------------------------
__global__ void k_final(const float* __restrict__ y0, const float* __restrict__ env,
                        const float* __restrict__ par, float* __restrict__ out) {
  const size_t idx = (size_t)blockIdx.x * blockDim.x + threadIdx.x;
  if (idx >= (size_t)BSZ * CHS * T_LEN) return;
  const int b = (int)(idx / ((size_t)CHS * T_LEN));
  const int t = (int)(idx % T_LEN);
  const float e = env[(size_t)b * T_LEN + t];
  const float gain = fminf(1.0f, par[RP_LTL + b] / fmaxf(e, 1e-7f));
  out[idx] = y0[idx] * gain;
}

// ---------------------------------------------------------------------------
extern "C" void kernel_launch(void* const* d_in, const int* in_sizes, int n_in,
                              void* d_out, int out_size, void* d_ws, size_t ws_size,
                              hipStream_t stream) {
  (void)in_sizes; (void)n_in; (void)out_size; (void)ws_size;
  const float* x        = (const float*)d_in[0];
  const float* drive_db = (const float*)d_in[1];
  const float* dist_mix = (const float*)d_in[2];
  const float* low_cut  = (const float*)d_in[3];
  const float* high_cut = (const float*)d_in[4];
  const float* mb_mix   = (const float*)d_in[5];
  const float* cth      = (const float*)d_in[6];
  const float* cra      = (const float*)d_in[7];
  const float* eth      = (const float*)d_in[8];
  const float* era      = (const float*)d_in[9];
  const float* atm      = (const float*)d_in[10];
  const float* rtm      = (const float*)d_in[11];
  const float* lth      = (const float*)d_in[12];
  const float* lat      = (const float*)d_in[13];
  const float* lrt      = (const float*)d_in[14];
  const int*   sr       = (const int*)d_in[15];

  float* W   = (float*)d_ws;
  float* y0  = W + OFF_Y0;
  float* low = W + OFF_LOW;
  float* high= W + OFF_HIGH;
  float* ub  = W + OFF_U;
  float* sb  = W + OFF_S;
  float* gb  = W + OFF_G;
  float* det = W + OFF_DET;
  float* env = W + OFF_ENV;
  float* par = W + OFF_PAR;
  float* out = (float*)d_out;

  const size_t Nel = (size_t)BSZ * CHS * T_LEN;

  k_coeffs<<<1, 64, 0, stream>>>(drive_db, low_cut, high_cut, cth, cra, eth, era,
                                 atm, rtm, lth, lat, lrt, sr, par);
  k_dist<<<(unsigned)((Nel + 255) / 256), 256, 0, stream>>>(x, dist_mix, par, y0);
  k_fir<<<dim3(64, 16), 32, 0, stream>>>(y0, par, low, high, ub);
  k_scan<<<2, 32, 0, stream>>>(par, ub, sb);
  const size_t N2 = (size_t)NSEQ_F * T_LEN;
  k_combine<<<(unsigned)((N2 + 255) / 256), 256, 0, stream>>>(par, sb, low, high);
  const size_t N3 = (size_t)48 * T_LEN;
  k_detgain<<<(unsigned)((N3 + 255) / 256), 256, 0, stream>>>(y0, low, high, par, gb);
  k_smooth_comp<<<2, 32, 0, stream>>>(par, gb);
  const size_t N4 = (size_t)BSZ * T_LEN;
  k_mbmix<<<(unsigned)((N4 + 255) / 256), 256, 0, stream>>>(low, high, gb, mb_mix, y0, det);
  k_smooth_lim<<<1, 32, 0, stream>>>(par, det, env);
  k_final<<<(unsigned)((Nel + 255) / 256), 256, 0, stream>>>(y0, env, par, out);
}